// GraphOperation_31662498906802
// MI455X (gfx1250) — compile-verified
//
#include <hip/hip_runtime.h>
#include <math.h>

#define B_ 8
#define C_ 64
#define N_ 400

typedef __attribute__((ext_vector_type(2))) float v2f;
typedef __attribute__((ext_vector_type(8))) float v8f;

// ---------------------------------------------------------------------------
// Pass 1: A[b,k,n] = relu(tanh(max_c R[b,c,k,n]))
// One thread per 4 consecutive (k,n) flat elements; 64 strided float4 loads,
// fully coalesced across the wave (adjacent lanes -> adjacent 16B chunks).
// ---------------------------------------------------------------------------
__global__ void k_adj(const float* __restrict__ R, float* __restrict__ A) {
    const int per_b4 = (N_ * N_) / 4;                  // 40000 float4 per (b,c) plane
    int idx = blockIdx.x * blockDim.x + threadIdx.x;   // [0, B_*per_b4)
    if (idx >= B_ * per_b4) return;
    int b    = idx / per_b4;
    int rem4 = idx - b * per_b4;

    const float4* p = (const float4*)(R + (size_t)b * C_ * N_ * N_) + rem4;
    float4 mx = p[0];
    #pragma unroll 4
    for (int c = 1; c < C_; ++c) {
        float4 v = p[(size_t)c * per_b4];
        mx.x = fmaxf(mx.x, v.x);
        mx.y = fmaxf(mx.y, v.y);
        mx.z = fmaxf(mx.z, v.z);
        mx.w = fmaxf(mx.w, v.w);
    }
    // relu(tanh(m)) == tanh(relu(m))
    float4 a;
    a.x = tanhf(fmaxf(mx.x, 0.f));
    a.y = tanhf(fmaxf(mx.y, 0.f));
    a.z = tanhf(fmaxf(mx.z, 0.f));
    a.w = tanhf(fmaxf(mx.w, 0.f));
    ((float4*)A)[idx] = a;
}

// ---------------------------------------------------------------------------
// Pass 2: x[b,c,k] = sum_n R[b,c,n,k] * A[b,k,n]
// One thread per 4 consecutive k for one (b,c): the big R stream is read as
// coalesced float4 (contiguous over k across lanes); the 4 A rows per thread
// are small and L2-resident (A = 5.1 MB).
// ---------------------------------------------------------------------------
__global__ void k_contract(const float* __restrict__ R, const float* __restrict__ A,
                           float* __restrict__ x) {
    const int kg = N_ / 4;                              // 100 k-groups
    int idx = blockIdx.x * blockDim.x + threadIdx.x;    // [0, B_*C_*kg)
    if (idx >= B_ * C_ * kg) return;
    int k4 = (idx % kg) * 4;
    int bc = idx / kg;
    int b  = bc / C_;

    const float4* rp = (const float4*)(R + (size_t)bc * N_ * N_ + k4); // step N_/4 per n
    const float*  a0 = A + ((size_t)b * N_ + (k4 + 0)) * N_;
    const float*  a1 = a0 + N_;
    const float*  a2 = a1 + N_;
    const float*  a3 = a2 + N_;

    float4 acc = make_float4(0.f, 0.f, 0.f, 0.f);
    #pragma unroll 4
    for (int n = 0; n < N_; ++n) {
        float4 r = rp[(size_t)n * (N_ / 4)];
        acc.x = fmaf(r.x, a0[n], acc.x);
        acc.y = fmaf(r.y, a1[n], acc.y);
        acc.z = fmaf(r.z, a2[n], acc.z);
        acc.w = fmaf(r.w, a3[n], acc.w);
    }
    ((float4*)x)[idx] = acc;
}

// ---------------------------------------------------------------------------
// Pass 3: Fg[b,o,n] = sum_c W[o,c] * x[b,c,n] + bias[o]
// Dense GEMM per batch: (64x64) @ (64x400). v_wmma_f32_16x16x4_f32, one wave
// per 16x16 tile, K looped 64/4 = 16 times. Fragment layouts per ISA 7.12.2:
//   A 16x4 : lane l (0-15) = row M=l; VGPR0 = K0 (lo half) / K2 (hi half),
//            VGPR1 = K1 / K3.
//   B 4x16 : lane l = col N=l; VGPR0 = K0/K2, VGPR1 = K1/K3.
//   D 16x16: VGPR r, lanes 0-15 -> M=r, lanes 16-31 -> M=r+8; N=l.
// ---------------------------------------------------------------------------
__global__ void k_linear_wmma(const float* __restrict__ W, const float* __restrict__ bias,
                              const float* __restrict__ x, float* __restrict__ out) {
    const int tilesN = N_ / 16;                       // 25
    const int tilesO = C_ / 16;                       // 4
    int tile = blockIdx.x * (blockDim.x >> 5) + (threadIdx.x >> 5);
    int nT = tile % tilesN; tile /= tilesN;
    int oT = tile % tilesO;
    int b  = tile / tilesO;

    int lane = threadIdx.x & 31;
    int half = lane >> 4;        // 0: lanes 0-15, 1: lanes 16-31
    int l    = lane & 15;

    const float* Wrow = W + (oT * 16 + l) * C_;           // A-frag row M=l
    const float* xcol = x + (size_t)b * C_ * N_ + nT * 16 + l; // B-frag col N=l

    v8f acc = (v8f)(0.f);
    #pragma unroll
    for (int k0 = 0; k0 < C_; k0 += 4) {
        int kk = k0 + 2 * half;
        v2f a, bf;
        a.x  = Wrow[kk];
        a.y  = Wrow[kk + 1];
        bf.x = xcol[(size_t)(kk)     * N_];
        bf.y = xcol[(size_t)(kk + 1) * N_];
        // 8 args: (neg_a, A, neg_b, B, c_mod, C, reuse_a, reuse_b)
        acc = __builtin_amdgcn_wmma_f32_16x16x4_f32(
            false, a, false, bf, (short)0, acc, false, false);
    }

    #pragma unroll
    for (int r = 0; r < 8; ++r) {
        int orow = oT * 16 + r + 8 * half;
        out[((size_t)b * C_ + orow) * N_ + nT * 16 + l] = acc[r] + bias[orow];
    }
}

// ---------------------------------------------------------------------------
extern "C" void kernel_launch(void* const* d_in, const int* in_sizes, int n_in,
                              void* d_out, int out_size, void* d_ws, size_t ws_size,
                              hipStream_t stream) {
    const float* R    = (const float*)d_in[0];  // (B, C, N, N)
    const float* W    = (const float*)d_in[1];  // (C, C)
    const float* bias = (const float*)d_in[2];  // (C,)
    float* out = (float*)d_out;                 // (B, C, N, 1)

    float* A = (float*)d_ws;                                            // B*N*N floats
    float* x = (float*)((char*)d_ws + (size_t)B_ * N_ * N_ * sizeof(float)); // B*C*N floats

    {   // Pass 1
        int total = (B_ * N_ * N_) / 4;                  // 320000 threads
        k_adj<<<(total + 255) / 256, 256, 0, stream>>>(R, A);
    }
    {   // Pass 2
        int total = (B_ * C_ * N_) / 4;                  // 51200 threads
        k_contract<<<(total + 255) / 256, 256, 0, stream>>>(R, A, x);
    }
    {   // Pass 3: 8 batches * 4 o-tiles * 25 n-tiles = 800 tiles, 8 waves/block
        int tiles = B_ * (C_ / 16) * (N_ / 16);          // 800
        k_linear_wmma<<<tiles / 8, 256, 0, stream>>>(W, bias, x, out);
    }
}